// SphereConv_49168785604700
// MI455X (gfx1250) — compile-verified
//
#include <hip/hip_runtime.h>
#include <math.h>

typedef __attribute__((ext_vector_type(2))) float v2f;
typedef __attribute__((ext_vector_type(8))) float v8f;

#define B_  4
#define C_  32
#define L_  256
#define M_  256
#define F_  8
#define N_  64

// ---------------------------------------------------------------------------
// Kernel 1: interpolate weights N=64 -> L=256 and pack as (L, 16, 32):
//   row 0..7  = Wr[f][c] at position l
//   row 8..15 = Wi[f][c] at position l
// This is the packed A-matrix (16 rows x C) for the WMMA stage.
// ---------------------------------------------------------------------------
__global__ __launch_bounds__(256) void interp_pack_kernel(
    const float* __restrict__ w_real, const float* __restrict__ w_imag,
    float* __restrict__ wsW) {
  int idx = blockIdx.x * blockDim.x + threadIdx.x;   // L*16*32 = 131072
  if (idx >= L_ * 16 * C_) return;
  int c   = idx & 31;
  int row = (idx >> 5) & 15;
  int l   = idx >> 9;

  // t = l/(L-1)*(N-1), lo = clamp(floor(t),0,N-2), frac = t-lo
  float t  = (float)l * ((float)(N_ - 1) / (float)(L_ - 1));
  int lo   = (int)floorf(t);
  if (lo < 0) lo = 0;
  if (lo > N_ - 2) lo = N_ - 2;
  float frac = t - (float)lo;

  int f = row & 7;
  const float* w = (row < 8) ? w_real : w_imag;
  float v0 = w[(f * C_ + c) * N_ + lo];
  float v1 = w[(f * C_ + c) * N_ + lo + 1];
  wsW[idx] = v0 * (1.0f - frac) + v1 * frac;
}

// ---------------------------------------------------------------------------
// Kernel 2: per (b,l): D1 = A*Xr, D2 = A*Xi with A = [Wr; Wi] (16 x 32),
// via 8x V_WMMA_F32_16X16X4_F32 per chain. Combine across lane halves with
// shfl_xor(16): lanes 0-15 -> Yr, lanes 16-31 -> Yi. Scale, ReLU(real), store.
// Block = 256 threads (8 waves); wave w covers m-tiles w and w+8.
// ---------------------------------------------------------------------------
__global__ __launch_bounds__(256) void sphereconv_wmma_kernel(
    const float* __restrict__ x_real, const float* __restrict__ x_imag,
    const float* __restrict__ wsW, float* __restrict__ out) {
  const int blk  = blockIdx.x;          // B*L blocks
  const int b    = blk / L_;
  const int l    = blk - b * L_;
  const int lane = threadIdx.x & 31;
  const int wave = threadIdx.x >> 5;
  const int half = lane >> 4;           // 0: lanes 0-15, 1: lanes 16-31
  const int col  = lane & 15;           // N-column / A-row within half

  // ---- A fragments (16x4 per chunk, 8 chunks cover K=C=32) ----
  // A layout: lane holds row M=lane%16; VGPR0 -> K = k0 + 2*half,
  //           VGPR1 -> K = k0 + 2*half + 1  (consecutive -> one float2 load)
  const float* wbase = wsW + ((size_t)(l * 16 + col) * C_);
  v2f a[8];
#pragma unroll
  for (int k = 0; k < 8; ++k) {
    int c0 = k * 4 + half * 2;
    a[k] = v2f{wbase[c0], wbase[c0 + 1]};
  }

  const size_t LM  = (size_t)L_ * M_;   // stride between channels c
  const float scale = sqrtf(1.0f + (float)l) * (1.0f / (float)C_);
  const float sgn   = half ? 1.0f : -1.0f;

#pragma unroll
  for (int t = 0; t < 2; ++t) {
    const int m0 = (wave + t * 8) * 16;
    const float* xrb = x_real + ((size_t)(b * C_) * L_ + l) * M_ + m0 + col;
    const float* xib = x_imag + ((size_t)(b * C_) * L_ + l) * M_ + m0 + col;

    v8f d1 = {};  // A * Xr : rows 0-7 = Wr*Xr, rows 8-15 = Wi*Xr
    v8f d2 = {};  // A * Xi : rows 0-7 = Wr*Xi, rows 8-15 = Wi*Xi
#pragma unroll
    for (int k = 0; k < 8; ++k) {
      // B layout (4x16): lane holds col N=lane%16; VGPR0 -> K = k0 + 2*half,
      //                  VGPR1 -> K = k0 + 2*half + 1
      int c0 = k * 4 + half * 2;
      v2f br = v2f{xrb[(size_t)c0 * LM], xrb[(size_t)(c0 + 1) * LM]};
      v2f bi = v2f{xib[(size_t)c0 * LM], xib[(size_t)(c0 + 1) * LM]};
      d1 = __builtin_amdgcn_wmma_f32_16x16x4_f32(
          false, a[k], false, br, (short)0, d1, false, false);
      d2 = __builtin_amdgcn_wmma_f32_16x16x4_f32(
          false, a[k], false, bi, (short)0, d2, false, false);
    }

    // Combine halves: VGPR v holds row v (lanes 0-15) / row v+8 (lanes 16-31).
    // lanes 0-15 : Yr[v] = WrXr - WiXi = d1 - shfl16(d2)
    // lanes 16-31: Yi[v] = WiXr + WrXi = d1 + shfl16(d2)
    const int comp = half;  // 0 = real plane, 1 = imag plane of output
    float* obase = out +
        ((((size_t)comp * B_ + b) * F_) * L_ + l) * M_ + m0 + col;
#pragma unroll
    for (int v = 0; v < 8; ++v) {
      float e2 = __shfl_xor(d2[v], 16, 32);
      float r  = (d1[v] + sgn * e2) * scale;
      r = half ? r : fmaxf(r, 0.0f);      // ReLU only on the real part
      obase[(size_t)v * LM] = r;          // f-stride = L*M
    }
  }
}

// ---------------------------------------------------------------------------
extern "C" void kernel_launch(void* const* d_in, const int* in_sizes, int n_in,
                              void* d_out, int out_size, void* d_ws, size_t ws_size,
                              hipStream_t stream) {
  (void)in_sizes; (void)n_in; (void)out_size; (void)ws_size;
  const float* x_real = (const float*)d_in[0];
  const float* x_imag = (const float*)d_in[1];
  const float* w_real = (const float*)d_in[2];
  const float* w_imag = (const float*)d_in[3];
  float* out = (float*)d_out;
  float* wsW = (float*)d_ws;              // (L,16,32) packed interp weights

  const int interpElems = L_ * 16 * C_;   // 131072
  interp_pack_kernel<<<(interpElems + 255) / 256, 256, 0, stream>>>(
      w_real, w_imag, wsW);

  sphereconv_wmma_kernel<<<B_ * L_, 256, 0, stream>>>(
      x_real, x_imag, wsW, out);
}